// InteractionHead_32770600469189
// MI455X (gfx1250) — compile-verified
//
#include <hip/hip_runtime.h>

// ---------------------------------------------------------------------------
// Problem constants (from reference)
// ---------------------------------------------------------------------------
constexpr int BIMG = 16;     // batch
constexpr int NBOX = 2048;   // proposals per image
constexpr int NCLS = 81;     // 80 fg + bg
constexpr int NFG  = 80;
constexpr int NGT  = 16;     // gt boxes per image
constexpr float SCORE_THRESH = 0.05f;
constexpr float NMS_THRESH   = 0.5f;
constexpr float FG_IOU_THRESH = 0.5f;
constexpr int MAX_HUMAN  = 10;
constexpr int MAX_OBJECT = 10;
constexpr int HUMAN_IDX  = 0;
constexpr int BG_CLASS   = 80;
constexpr int NCHUNK = NBOX / 16;   // 128 16-col chunks per row
constexpr int CSUB   = 8;           // column sub-tiles per wave (16x128 strip)

typedef float v2f __attribute__((ext_vector_type(2)));
typedef float v8f __attribute__((ext_vector_type(8)));

// ---------------------------------------------------------------------------
// K0: zero the tiny accumulators (boxmax per image as int-bits, 2 loss accums)
// ---------------------------------------------------------------------------
__global__ void k_init(int* boxmax, float* accum) {
  int t = threadIdx.x;
  if (t < BIMG) boxmax[t] = 0;          // all coords >= 0, int-bit max works
  if (t < 2)    accum[t] = 0.0f;
}

// ---------------------------------------------------------------------------
// K1: per-proposal softmax score / label / sort key; per-image box-coord max
// key = [orderable(score'), 32 bits][~index, 32 bits]  -> descending sort
// gives jnp.argsort(-score) incl. stable tie-break on smaller index.
// ---------------------------------------------------------------------------
__global__ __launch_bounds__(256)
void k_scores(const float* __restrict__ boxes, const float* __restrict__ logits,
              unsigned long long* __restrict__ keys, int* __restrict__ labels,
              int* __restrict__ boxmax) {
  int gid = blockIdx.x * 256 + threadIdx.x;     // one (b,n) per thread
  int b = gid / NBOX;
  int n = gid - b * NBOX;
  const float* l = logits + (size_t)gid * NCLS;

  float m = -1e30f;
  #pragma unroll 3
  for (int c = 0; c < NCLS; ++c) m = fmaxf(m, l[c]);
  float s = 0.0f;
  #pragma unroll 3
  for (int c = 0; c < NCLS; ++c) s += __expf(l[c] - m);
  float best = -1e30f; int lab = 0;
  for (int c = 0; c < NFG; ++c) {
    float v = l[c];
    if (v > best) { best = v; lab = c; }        // strict > => first argmax
  }
  float score = __expf(best - m) / s;
  bool act = (score >= SCORE_THRESH);
  float sp = act ? score : -1.0f;
  unsigned u = __float_as_uint(sp);
  u = (u >> 31) ? ~u : (u | 0x80000000u);       // order-preserving map
  keys[gid]   = ((unsigned long long)u << 32) | (unsigned)(0xFFFFFFFFu - (unsigned)n);
  labels[gid] = lab;

  // per-image max over all 4 coords -> boxmax[b]
  const float* bx = boxes + (size_t)gid * 4;
  float mx = fmaxf(fmaxf(bx[0], bx[1]), fmaxf(bx[2], bx[3]));
  __shared__ float red[256];
  red[threadIdx.x] = mx;
  __syncthreads();
  for (int off = 128; off > 0; off >>= 1) {
    if (threadIdx.x < off) red[threadIdx.x] = fmaxf(red[threadIdx.x], red[threadIdx.x + off]);
    __syncthreads();
  }
  if (threadIdx.x == 0) atomicMax(boxmax + b, __float_as_int(red[0]));
}

// ---------------------------------------------------------------------------
// K2: per-image LDS bitonic sort (descending) of 2048 64-bit keys, then
// gather order / active / labels / boxes into sorted arrays.
// ---------------------------------------------------------------------------
__global__ __launch_bounds__(1024)
void k_sort(const float* __restrict__ boxes, const unsigned long long* __restrict__ keys,
            const int* __restrict__ labels, int* __restrict__ order,
            int* __restrict__ act_sorted, int* __restrict__ labels_sorted,
            float* __restrict__ boxes_sorted) {
  __shared__ unsigned long long s[NBOX];
  const int b = blockIdx.x;
  const int t = threadIdx.x;
  const size_t base = (size_t)b * NBOX;
  s[t]        = keys[base + t];
  s[t + 1024] = keys[base + t + 1024];
  __syncthreads();
  for (int k = 2; k <= NBOX; k <<= 1) {
    for (int j = k >> 1; j > 0; j >>= 1) {
      #pragma unroll
      for (int p = 0; p < 2; ++p) {
        int idx = t + p * 1024;
        int ixj = idx ^ j;
        if (ixj > idx) {
          unsigned long long a = s[idx], c = s[ixj];
          bool sw = ((idx & k) == 0) ? (a < c) : (a > c);   // descending net
          if (sw) { s[idx] = c; s[ixj] = a; }
        }
      }
      __syncthreads();
    }
  }
  #pragma unroll
  for (int p = 0; p < 2; ++p) {
    int i = t + p * 1024;
    unsigned long long key = s[i];
    unsigned idx32 = 0xFFFFFFFFu - (unsigned)(key & 0xFFFFFFFFu);
    unsigned hi = (unsigned)(key >> 32);
    unsigned ob = (hi & 0x80000000u) ? (hi ^ 0x80000000u) : ~hi;
    float score = __uint_as_float(ob);
    order[base + i]         = (int)idx32;
    act_sorted[base + i]    = (score >= SCORE_THRESH) ? 1 : 0;
    labels_sorted[base + i] = labels[base + idx32];
    ((float4*)boxes_sorted)[base + i] = ((const float4*)boxes)[base + idx32];
  }
}

// ---------------------------------------------------------------------------
// K3: suppression bitmask. One wave32 per 16x128 (row x col) strip of sorted,
// class-offset boxes (8 sub-tiles of 16x16).  Pairwise union denominators
// area_i+area_j are a rank-2 outer product -> one V_WMMA_F32_16X16X4_F32 per
// 16x16 sub-tile, with the A fragment (row areas) built once per strip:
//   A(16x4): col0 = area_row, col1 = 1      (rows M=0..15)
//   B(4x16): row0 = 1,        row1 = area_col
//   D(16x16) = area_row[M] + area_col[N]
// The NMS predicate inter/(union+eps) > 0.5 is evaluated division-free as
// inter > 0.5*(sum - inter + eps)  (denominator strictly positive).
// Each sub-tile exclusively owns mask16[row][ctile]; lower-triangle sub-tiles
// write 0 so no memset of the 8MB mask is needed.
// ---------------------------------------------------------------------------
__global__ __launch_bounds__(32)
void k_nmsmask(const float* __restrict__ boxes_sorted, const int* __restrict__ labels_sorted,
               const int* __restrict__ boxmax, unsigned short* __restrict__ mask16) {
  const int cgroup = blockIdx.x, rtile = blockIdx.y, b = blockIdx.z;
  const int lane = threadIdx.x;
  const bool lo = (lane < 16);
  const int rbase = rtile * 16;
  const size_t base = (size_t)b * NBOX;
  const float offscale = __int_as_float(boxmax[b]) + 1.0f;

  // --- row side: load once per strip ---
  __shared__ float rbox[16][4];
  float rowArea = 0.0f;
  if (lo) {
    int r = rbase + lane;
    float4 bb = ((const float4*)boxes_sorted)[base + r];
    float off = (float)labels_sorted[base + r] * offscale;
    rbox[lane][0] = bb.x + off; rbox[lane][1] = bb.y + off;
    rbox[lane][2] = bb.z + off; rbox[lane][3] = bb.w + off;
    rowArea = (bb.z - bb.x) * (bb.w - bb.y);
  }
  __syncthreads();

  // per-lane private copy of its 8 row boxes (rloc = v + 8*hi-half)
  float rx0[8], ry0[8], rx1[8], ry1[8];
  #pragma unroll
  for (int v = 0; v < 8; ++v) {
    int rloc = v + (lo ? 0 : 8);
    rx0[v] = rbox[rloc][0]; ry0[v] = rbox[rloc][1];
    rx1[v] = rbox[rloc][2]; ry1[v] = rbox[rloc][3];
  }

  // WMMA A fragment: invariant across the strip
  v2f a;
  a[0] = lo ? rowArea : 0.0f;
  a[1] = lo ? 1.0f    : 0.0f;
  const v8f cz = {0.f, 0.f, 0.f, 0.f, 0.f, 0.f, 0.f, 0.f};

  for (int sub = 0; sub < CSUB; ++sub) {
    const int ctile = cgroup * CSUB + sub;
    if (ctile < rtile) {                     // j > i never holds: write zeros
      if (lo) mask16[(base + rbase + lane) * NCHUNK + ctile] = 0;
      continue;
    }
    const int cbase = ctile * 16;
    const int col = cbase + (lane & 15);
    float4 cb = ((const float4*)boxes_sorted)[base + col];
    float coff = (float)labels_sorted[base + col] * offscale;
    const float cx0 = cb.x + coff, cy0 = cb.y + coff;
    const float cx1 = cb.z + coff, cy1 = cb.w + coff;
    const float colArea = (cb.z - cb.x) * (cb.w - cb.y);

    v2f bf;
    bf[0] = lo ? 1.0f    : 0.0f;
    bf[1] = lo ? colArea : 0.0f;
    v8f sums = __builtin_amdgcn_wmma_f32_16x16x4_f32(false, a, false, bf,
                                                     (short)0, cz, false, false);

    #pragma unroll
    for (int v = 0; v < 8; ++v) {
      int row = rbase + v + (lo ? 0 : 8);
      float iw = fmaxf(fminf(rx1[v], cx1) - fmaxf(rx0[v], cx0), 0.0f);
      float ih = fmaxf(fminf(ry1[v], cy1) - fmaxf(ry0[v], cy0), 0.0f);
      float inter = iw * ih;
      // inter/(sum - inter + eps) > 0.5  <=>  inter > 0.5*(sum - inter + eps)
      bool pred = (inter > NMS_THRESH * (sums[v] - inter + 1e-9f)) && (col > row);
      unsigned bal = __builtin_amdgcn_ballot_w32(pred);
      if (lane == 0)
        mask16[(base + rbase + v) * NCHUNK + ctile] = (unsigned short)(bal & 0xFFFFu);
      if (lane == 16)
        mask16[(base + rbase + v + 8) * NCHUNK + ctile] = (unsigned short)((bal >> 16) & 0xFFFFu);
    }
  }
}

// ---------------------------------------------------------------------------
// K4: per-image serial greedy NMS over the bitmask (64-word LDS bitset),
// then block-wide scan for the human/object caps -> final keep flags.
// ---------------------------------------------------------------------------
__global__ __launch_bounds__(256)
void k_nms_seq(const unsigned short* __restrict__ mask16, const int* __restrict__ act_sorted,
               const int* __restrict__ labels_sorted, int* __restrict__ finalKeep) {
  const int b = blockIdx.x;
  const int t = threadIdx.x;
  const size_t base = (size_t)b * NBOX;
  __shared__ unsigned sup[NBOX / 32];
  __shared__ int scanH[256], scanO[256];

  if (t < 64) {
    unsigned w = 0;
    for (int k = 0; k < 32; ++k)
      if (!act_sorted[base + t * 32 + k]) w |= (1u << k);   // sup init = !act
    sup[t] = w;
  }
  __syncthreads();

  for (int i = 0; i < NBOX; ++i) {
    bool supd = (sup[i >> 5] >> (i & 31)) & 1u;
    __syncthreads();
    if (!supd && t < 64) {
      const unsigned short* mrow = mask16 + (base + i) * NCHUNK;
      sup[t] |= (unsigned)mrow[2 * t] | ((unsigned)mrow[2 * t + 1] << 16);
    }
    __syncthreads();
  }

  int hloc[8], oloc[8], th = 0, to = 0;
  #pragma unroll
  for (int k = 0; k < 8; ++k) {
    int i = t * 8 + k;
    bool keep = !((sup[i >> 5] >> (i & 31)) & 1u);
    int lab = labels_sorted[base + i];
    int h = (keep && lab == HUMAN_IDX) ? 1 : 0;
    int o = (keep && lab != HUMAN_IDX) ? 1 : 0;
    th += h; to += o; hloc[k] = h; oloc[k] = o;
  }
  scanH[t] = th; scanO[t] = to;
  __syncthreads();
  for (int off = 1; off < 256; off <<= 1) {
    int vh = (t >= off) ? scanH[t - off] : 0;
    int vo = (t >= off) ? scanO[t - off] : 0;
    __syncthreads();
    scanH[t] += vh; scanO[t] += vo;
    __syncthreads();
  }
  int ch = scanH[t] - th, co = scanO[t] - to;
  #pragma unroll
  for (int k = 0; k < 8; ++k) {
    int i = t * 8 + k;
    ch += hloc[k]; co += oloc[k];
    int hk = (hloc[k] && ch <= MAX_HUMAN) ? 1 : 0;
    int ok = (oloc[k] && co <= MAX_OBJECT) ? 1 : 0;
    finalKeep[base + i] = hk | ok;
  }
}

// ---------------------------------------------------------------------------
// K5: GT assignment (N x 16 IoU, LDS-staged GT, division-free cross-product
// argmax), log-softmax CE, block reduction -> global accumulators.
// ---------------------------------------------------------------------------
__global__ __launch_bounds__(256)
void k_ce(const float* __restrict__ logits, const float* __restrict__ gt_boxes,
          const int* __restrict__ gt_classes, const float* __restrict__ boxes_sorted,
          const int* __restrict__ order, const int* __restrict__ finalKeep,
          float* __restrict__ accum) {
  const int gid = blockIdx.x * 256 + threadIdx.x;
  const int b = gid / NBOX;
  const int i = gid - b * NBOX;
  const size_t base = (size_t)b * NBOX;

  __shared__ float4 gbox[NGT];
  __shared__ float  garea[NGT];
  __shared__ int    gcls[NGT];
  if (threadIdx.x < NGT) {
    float4 g = ((const float4*)gt_boxes)[(size_t)b * NGT + threadIdx.x];
    gbox[threadIdx.x] = g;
    garea[threadIdx.x] = (g.z - g.x) * (g.w - g.y);
    gcls[threadIdx.x] = gt_classes[b * NGT + threadIdx.x];
  }
  __syncthreads();

  const int o = order[base + i];
  const float* l = logits + (base + o) * NCLS;
  __builtin_prefetch(l, 0, 0);                 // global_prefetch_b8

  float4 bb = ((const float4*)boxes_sorted)[base + i];
  float areaA = (bb.z - bb.x) * (bb.w - bb.y);
  // argmax of num_g/den_g via cross products (den > 0); first max kept
  float bestNum = -1.0f, bestDen = 1.0f; int barg = 0;
  #pragma unroll
  for (int g = 0; g < NGT; ++g) {
    float4 gb = gbox[g];
    float iw = fmaxf(fminf(bb.z, gb.z) - fmaxf(bb.x, gb.x), 0.0f);
    float ih = fmaxf(fminf(bb.w, gb.w) - fmaxf(bb.y, gb.y), 0.0f);
    float num = iw * ih;
    float den = areaA + garea[g] - num + 1e-9f;
    if (num * bestDen > bestNum * den) { bestNum = num; bestDen = den; barg = g; }
  }
  int assigned = (bestNum >= FG_IOU_THRESH * bestDen) ? gcls[barg] : BG_CLASS;

  float m = -1e30f;
  #pragma unroll 3
  for (int c = 0; c < NCLS; ++c) m = fmaxf(m, l[c]);
  float s = 0.0f;
  #pragma unroll 3
  for (int c = 0; c < NCLS; ++c) s += __expf(l[c] - m);
  float ce = -(l[assigned] - m - __logf(s));

  int f = finalKeep[base + i];
  __shared__ float red[256];
  red[threadIdx.x] = f ? ce : 0.0f;
  __syncthreads();
  for (int off = 128; off > 0; off >>= 1) {
    if (threadIdx.x < off) red[threadIdx.x] += red[threadIdx.x + off];
    __syncthreads();
  }
  if (threadIdx.x == 0) atomicAdd(accum + 0, red[0]);
  __syncthreads();
  red[threadIdx.x] = (float)f;
  __syncthreads();
  for (int off = 128; off > 0; off >>= 1) {
    if (threadIdx.x < off) red[threadIdx.x] += red[threadIdx.x + off];
    __syncthreads();
  }
  if (threadIdx.x == 0) atomicAdd(accum + 1, red[0]);
}

// ---------------------------------------------------------------------------
// K6: finalize scalar
// ---------------------------------------------------------------------------
__global__ void k_fin(const float* __restrict__ accum, float* __restrict__ out) {
  if (threadIdx.x == 0) out[0] = accum[0] / fmaxf(accum[1], 1.0f);
}

// ---------------------------------------------------------------------------
extern "C" void kernel_launch(void* const* d_in, const int* in_sizes, int n_in,
                              void* d_out, int out_size, void* d_ws, size_t ws_size,
                              hipStream_t stream) {
  const float* boxes      = (const float*)d_in[0];   // (16,2048,4)
  const float* logits     = (const float*)d_in[1];   // (16,2048,81)
  const float* gt_boxes   = (const float*)d_in[2];   // (16,16,4)
  const int*   gt_classes = (const int*)d_in[3];     // (16,16)
  float* out = (float*)d_out;

  // workspace carve-up (all 256B aligned)
  char* ws = (char*)d_ws;
  size_t off = 0;
  auto carve = [&](size_t bytes) { void* p = ws + off; off = (off + bytes + 255) & ~(size_t)255; return p; };
  unsigned long long* keys = (unsigned long long*)carve((size_t)BIMG * NBOX * 8);
  int*   labels        = (int*)  carve((size_t)BIMG * NBOX * 4);
  int*   order         = (int*)  carve((size_t)BIMG * NBOX * 4);
  int*   act_sorted    = (int*)  carve((size_t)BIMG * NBOX * 4);
  int*   labels_sorted = (int*)  carve((size_t)BIMG * NBOX * 4);
  float* boxes_sorted  = (float*)carve((size_t)BIMG * NBOX * 16);
  int*   boxmax        = (int*)  carve((size_t)BIMG * 4);
  float* accum         = (float*)carve(2 * 4);
  int*   finalKeep     = (int*)  carve((size_t)BIMG * NBOX * 4);
  unsigned short* mask16 = (unsigned short*)carve((size_t)BIMG * NBOX * NCHUNK * 2);
  (void)ws_size; (void)in_sizes; (void)n_in; (void)out_size;

  k_init<<<1, 32, 0, stream>>>(boxmax, accum);
  k_scores<<<(BIMG * NBOX) / 256, 256, 0, stream>>>(boxes, logits, keys, labels, boxmax);
  k_sort<<<BIMG, 1024, 0, stream>>>(boxes, keys, labels, order, act_sorted,
                                    labels_sorted, boxes_sorted);
  k_nmsmask<<<dim3(NCHUNK / CSUB, NCHUNK, BIMG), 32, 0, stream>>>(boxes_sorted, labels_sorted,
                                                                  boxmax, mask16);
  k_nms_seq<<<BIMG, 256, 0, stream>>>(mask16, act_sorted, labels_sorted, finalKeep);
  k_ce<<<(BIMG * NBOX) / 256, 256, 0, stream>>>(logits, gt_boxes, gt_classes,
                                                boxes_sorted, order, finalKeep, accum);
  k_fin<<<1, 32, 0, stream>>>(accum, out);
}